// GradICON_43619687858281
// MI455X (gfx1250) — compile-verified
//
#include <hip/hip_runtime.h>

// ---------------------------------------------------------------------------
// GradICON forward, fused. B=8, C=1, H=W=1024.
// SPACING = 1/1023 (align_corners), Ieps noise scale = 1/1024, DELTA=1e-3.
// Per-pixel gather work + WMMA-based (V_WMMA_F32_16X16X4_F32) exact f32
// cross-lane reductions, LDS cross-wave reduction, atomics into d_ws.
// ---------------------------------------------------------------------------

typedef float v2f __attribute__((ext_vector_type(2)));
typedef float v8f __attribute__((ext_vector_type(8)));

#define HW (1024 * 1024)
#define INV_S 1023.0f      // 1/SPACING
#define SPC (1.0f / 1023.0f)
#define NOISE_SC (1.0f / 1024.0f)
#define DLT 0.001f
#define INV_DLT 1000.0f

// Exact 32-lane f32 sum using WMMA f32_16x16x4_f32 with an all-ones B matrix.
// A layout (16x4 f32): lane m (m<16) holds A[m][0] (vgpr0), A[m][1] (vgpr1);
// lane m+16 holds A[m][2], A[m][3]. With a[1]=0 and B=ones:
//   D[m][n] = v_m + v_{m+16}  (independent of n, exact f32 adds).
// D layout: lane n<16 vgpr r = D[r][n]; lane n>=16 vgpr r = D[r+8][n-16].
// Sum of 8 D regs gives half-sums; shfl_xor(16) completes the total in all lanes.
__device__ __forceinline__ float wave_sum(float v) {
    v2f a; a[0] = v;    a[1] = 0.0f;
    v2f b; b[0] = 1.0f; b[1] = 1.0f;
    v8f c = {};
    v8f d = __builtin_amdgcn_wmma_f32_16x16x4_f32(false, a, false, b,
                                                  (short)0, c, false, false);
    float s = ((d[0] + d[1]) + (d[2] + d[3])) + ((d[4] + d[5]) + (d[6] + d[7]));
    s += __shfl_xor(s, 16, 32);
    return s;
}

// Bilinear sample of a 2-channel field (channel pointers f0,f1; row-major
// 1024x1024, row stride 1024) at physical coords (phix,phiy).
// Border clamp + align_corners, matching the reference warp() exactly.
__device__ __forceinline__ void sample2(const float* __restrict__ f0,
                                        const float* __restrict__ f1,
                                        float phix, float phiy,
                                        float& ox, float& oy) {
    float px = fminf(fmaxf(phix * INV_S, 0.0f), 1023.0f);
    float py = fminf(fmaxf(phiy * INV_S, 0.0f), 1023.0f);
    int x0 = (int)px; x0 = x0 > 1022 ? 1022 : x0;
    int y0 = (int)py; y0 = y0 > 1022 ? 1022 : y0;
    float wx = px - (float)x0;
    float wy = py - (float)y0;
    int base = (x0 << 10) + y0;
    float a00 = f0[base],        a01 = f0[base + 1];
    float a10 = f0[base + 1024], a11 = f0[base + 1025];
    float b00 = f1[base],        b01 = f1[base + 1];
    float b10 = f1[base + 1024], b11 = f1[base + 1025];
    float iwx = 1.0f - wx, iwy = 1.0f - wy;
    ox = iwx * (iwy * a00 + wy * a01) + wx * (iwy * a10 + wy * a11);
    oy = iwx * (iwy * b00 + wy * b01) + wx * (iwy * b10 + wy * b11);
}

// Bilinear sample of an image + the analytic interior tag (1 on [1,1022]^2)
// with identical taps/weights (reference warps concat([img, tag])).
__device__ __forceinline__ void sample_img_mask(const float* __restrict__ img,
                                                float phix, float phiy,
                                                float& val, float& mask) {
    float px = fminf(fmaxf(phix * INV_S, 0.0f), 1023.0f);
    float py = fminf(fmaxf(phiy * INV_S, 0.0f), 1023.0f);
    int x0 = (int)px; x0 = x0 > 1022 ? 1022 : x0;
    int y0 = (int)py; y0 = y0 > 1022 ? 1022 : y0;
    float wx = px - (float)x0;
    float wy = py - (float)y0;
    int base = (x0 << 10) + y0;
    float v00 = img[base],        v01 = img[base + 1];
    float v10 = img[base + 1024], v11 = img[base + 1025];
    float r0 = (x0 >= 1)    ? 1.0f : 0.0f;   // row x0   interior (x0<=1022 always)
    float r1 = (x0 <= 1021) ? 1.0f : 0.0f;   // row x0+1 interior
    float c0 = (y0 >= 1)    ? 1.0f : 0.0f;
    float c1 = (y0 <= 1021) ? 1.0f : 0.0f;
    float iwx = 1.0f - wx, iwy = 1.0f - wy;
    val  = iwx * (iwy * v00       + wy * v01)       + wx * (iwy * v10       + wy * v11);
    mask = iwx * (iwy * (r0 * c0) + wy * (r0 * c1)) + wx * (iwy * (r1 * c0) + wy * (r1 * c1));
}

__global__ void gradicon_zero(float* __restrict__ ws) {
    int t = threadIdx.x;
    if (t < 35) ws[t] = 0.0f;
}

// ws layout: [0]=icon_sum [1]=tm_sum [2]=flips
//            [3+b]=numA [11+b]=denA [19+b]=numB [27+b]=denB
__global__ __launch_bounds__(256) void gradicon_main(
    const float* __restrict__ imgA, const float* __restrict__ imgB,
    const float* __restrict__ dAB,  const float* __restrict__ dBA,
    const float* __restrict__ noise, float* __restrict__ ws) {

    const int bid = blockIdx.x;          // ((b*1024 + i)*4 + jb)
    const int b   = bid >> 12;
    const int i   = (bid >> 2) & 1023;
    const int j   = ((bid & 3) << 8) + threadIdx.x;
    const int pix = (i << 10) + j;

    const float* A   = imgA  + b * HW;
    const float* Bi  = imgB  + b * HW;
    const float* ab0 = dAB   + b * 2 * HW;  const float* ab1 = ab0 + HW;
    const float* ba0 = dBA   + b * 2 * HW;  const float* ba1 = ba0 + HW;
    const float* n0  = noise + b * 2 * HW;  const float* n1  = n0 + HW;

    const float gx = (float)i * SPC;
    const float gy = (float)j * SPC;

    // ---------------- icon loss ----------------
    const float xx = gx + n0[pix] * NOISE_SC;
    const float xy = gy + n1[pix] * NOISE_SC;

    auto compose = [&](float px, float py, float& ox, float& oy) {
        float ux, uy;
        sample2(ba0, ba1, px, py, ux, uy);       // warp(disp_BA, x)
        float yx = px + ux, yy = py + uy;        // phi_BA(x)
        float vx, vy;
        sample2(ab0, ab1, yx, yy, vx, vy);       // warp(disp_AB, y)
        ox = yx + vx; oy = yy + vy;              // phi_AB(phi_BA(x))
    };

    float c0x, c0y; compose(xx, xy, c0x, c0y);
    const float e0x = xx - c0x, e0y = xy - c0y;

    float cx, cy;
    compose(xx + DLT, xy, cx, cy);
    float edx = (xx + DLT) - cx, edy = xy - cy;
    float tx = (e0x - edx) * INV_DLT, ty = (e0y - edy) * INV_DLT;
    float icon = tx * tx + ty * ty;

    compose(xx, xy + DLT, cx, cy);
    edx = xx - cx; edy = (xy + DLT) - cy;
    tx = (e0x - edx) * INV_DLT; ty = (e0y - edy) * INV_DLT;
    icon += tx * tx + ty * ty;

    // ---- direct reads: phi_*_vf = idm + disp (idm lands on grid exactly) ----
    const float dabx = ab0[pix], daby = ab1[pix];
    const float dbax = ba0[pix], dbay = ba1[pix];

    // transform magnitude: mean((idm - phi_AB_vf)^2) = mean(disp_AB^2)
    const float tm = dabx * dabx + daby * daby;

    // ---------------- similarity ----------------
    float wA, mA;
    sample_img_mask(A, gx + dabx, gy + daby, wA, mA);
    float dfa = wA - Bi[pix];
    const float numA = mA * dfa * dfa, denA = mA;

    float wB, mB;
    sample_img_mask(Bi, gx + dbax, gy + dbay, wB, mB);
    float dfb = wB - A[pix];
    const float numB = mB * dfb * dfb, denB = mB;

    // ---------------- flips on phi_BA_vf ----------------
    float fl = 0.0f;
    {
        const bool valid = (i < 1023) && (j < 1023);
        const int pR = (i < 1023) ? pix + 1024 : pix;  // (i+1, j), clamped load
        const int pC = (j < 1023) ? pix + 1    : pix;  // (i, j+1), clamped load
        float du0 = SPC + (ba0[pR] - dbax);
        float du1 =        ba1[pR] - dbay;
        float dv0 =        ba0[pC] - dbax;
        float dv1 = SPC + (ba1[pC] - dbay);
        float dA  = du0 * dv1 - du1 * dv0;
        fl = (valid && (dA < 0.0f)) ? 1.0f : 0.0f;
    }

    // ---------------- reductions (WMMA wave-sum, uniform control flow) -------
    float r0 = wave_sum(icon);
    float r1 = wave_sum(tm);
    float r2 = wave_sum(fl);
    float r3 = wave_sum(numA);
    float r4 = wave_sum(denA);
    float r5 = wave_sum(numB);
    float r6 = wave_sum(denB);

    __shared__ float sm[8][8];
    const int wv = threadIdx.x >> 5;
    const int ln = threadIdx.x & 31;
    if (ln == 0) {
        sm[wv][0] = r0; sm[wv][1] = r1; sm[wv][2] = r2; sm[wv][3] = r3;
        sm[wv][4] = r4; sm[wv][5] = r5; sm[wv][6] = r6;
    }
    __syncthreads();

    const int t = threadIdx.x;
    if (t < 7) {
        float s = 0.0f;
#pragma unroll
        for (int w = 0; w < 8; ++w) s += sm[w][t];
        int tgt;
        switch (t) {
            case 0: tgt = 0; break;
            case 1: tgt = 1; break;
            case 2: tgt = 2; break;
            case 3: tgt = 3 + b; break;
            case 4: tgt = 11 + b; break;
            case 5: tgt = 19 + b; break;
            default: tgt = 27 + b; break;
        }
        atomicAdd(&ws[tgt], s);
    }
}

__global__ void gradicon_final(const float* __restrict__ ws,
                               float* __restrict__ out) {
    if (threadIdx.x == 0 && blockIdx.x == 0) {
        const float inv_n = 1.0f / 16777216.0f;   // B*2*H*W
        float icon = ws[0] * inv_n;
        float tm   = ws[1] * inv_n;
        float fl   = ws[2] * 0.125f;              // /B
        float sim  = 0.0f;
#pragma unroll
        for (int b = 0; b < 8; ++b) sim += ws[3 + b]  / ws[11 + b];
#pragma unroll
        for (int b = 0; b < 8; ++b) sim += ws[19 + b] / ws[27 + b];
        sim *= 0.125f;                            // mean over batch
        out[0] = 1.5f * icon + sim;               // all_loss (LMBDA=1.5)
        out[1] = icon;
        out[2] = sim;
        out[3] = tm;
        out[4] = fl;
    }
}

extern "C" void kernel_launch(void* const* d_in, const int* in_sizes, int n_in,
                              void* d_out, int out_size, void* d_ws, size_t ws_size,
                              hipStream_t stream) {
    (void)in_sizes; (void)n_in; (void)out_size; (void)ws_size;
    const float* imgA  = (const float*)d_in[0];
    const float* imgB  = (const float*)d_in[1];
    const float* dAB   = (const float*)d_in[2];
    const float* dBA   = (const float*)d_in[3];
    const float* noise = (const float*)d_in[4];
    float* out = (float*)d_out;
    float* ws  = (float*)d_ws;

    gradicon_zero<<<1, 64, 0, stream>>>(ws);
    gradicon_main<<<32768, 256, 0, stream>>>(imgA, imgB, dAB, dBA, noise, ws);
    gradicon_final<<<1, 32, 0, stream>>>(ws, out);
}